// PHGATLayer_64725157151124
// MI455X (gfx1250) — compile-verified
//
#include <hip/hip_runtime.h>
#include <math.h>

// ---------------------------------------------------------------------------
// PHGAT layer for MI455X (gfx1250, wave32).
//  - Dense projections: V_WMMA_F32_16X16X4_F32 (native f32 matrix path,
//    preserves reference fp32 semantics). 13 GEMMs [M,256]x[256,128].
//    B panel staged into LDS per K-chunk via GLOBAL_LOAD_ASYNC_TO_LDS_B128
//    (ASYNCcnt-tracked CDNA5 async copy), WMMAs read B via ds_load.
//  - Edge phase: one wave per edge; float4 row slices per lane, shuffle
//    reduction for cosine score, global f32 atomics for segment sums.
//  - ht written directly into d_out[:, 0:128]; etypes 4..7 scatter messages
//    directly into d_out[:, 128:256] (their softmax weight is identically 1).
// ---------------------------------------------------------------------------

#define N_VUL   100000
#define N_OTH   20000
#define NE      400000
#define IN_DIM  256
#define OUT_DIM 128
#define KCHUNK  64                      // K rows of B staged per LDS buffer

typedef __attribute__((ext_vector_type(2))) float v2f;
typedef __attribute__((ext_vector_type(8))) float v8f;

__device__ __forceinline__ void wait_async0() {
#if __has_builtin(__builtin_amdgcn_s_wait_asynccnt)
  __builtin_amdgcn_s_wait_asynccnt(0);
#else
  asm volatile("s_wait_asynccnt 0x0" ::: "memory");
#endif
}

// ---------------- zero fill -------------------------------------------------
__global__ void __launch_bounds__(256)
zero_f32(float* __restrict__ p, size_t n) {
  size_t i = (size_t)blockIdx.x * blockDim.x + threadIdx.x;
  size_t stride = (size_t)gridDim.x * blockDim.x;
  for (; i < n; i += stride) p[i] = 0.0f;
}

// ---------------- WMMA f32 GEMM:  C[M,128] = A[M,256] * B[256,128] (+bias) --
// Block: 256 threads = 8 waves; wave w owns N-tile w (cols w*16..w*16+15).
// blockIdx.x owns rows m0..m0+15.  K-loop: 4 chunks x 16 WMMA.
// B chunk [KCHUNK,128] (32 KB) is async-copied to LDS once per block and
// shared by all 8 waves (kills the 8x-redundant VMEM fetch of B).
// A 16x4 lane layout (ISA 7.12.2): lanes 0-15 K={k,k+1}, lanes 16-31
// K={k+2,k+3}, row = lane%16.  B 4x16 mirrored: col = lane%16, same K split.
// D (v8f): row = r + 8*(lane/16), col = lane%16.
__global__ void __launch_bounds__(256)
wmma_gemm_f32(const float* __restrict__ A,
              const float* __restrict__ B,
              const float* __restrict__ bias,
              float* __restrict__ C, int ldc, int M)
{
  __shared__ float Bs[KCHUNK * OUT_DIM];          // 32 KB
  const int tid  = threadIdx.x;
  const int wave = tid >> 5;
  const int lane = tid & 31;
  const int half = lane >> 4;                     // 0: lanes 0-15, 1: 16-31
  const int l16  = lane & 15;
  const int m0   = blockIdx.x * 16;
  const int n0   = wave * 16;
  if (m0 >= M) return;

  v8f acc = {0.f, 0.f, 0.f, 0.f, 0.f, 0.f, 0.f, 0.f};
  const float* arow = A + (size_t)(m0 + l16) * IN_DIM + 2 * half;
  const float* bcol = Bs + (size_t)(2 * half) * OUT_DIM + n0 + l16;
  const unsigned lds_base = (unsigned)(unsigned long long)(uintptr_t)&Bs[tid * 4];

  for (int kc = 0; kc < IN_DIM; kc += KCHUNK) {
    __syncthreads();                              // previous chunk consumed
    {
      const float* gsrc = B + (size_t)kc * OUT_DIM + tid * 4;
#pragma unroll
      for (int i = 0; i < (KCHUNK * OUT_DIM) / (256 * 4); ++i) {  // 8 x 16B/lane
        asm volatile("global_load_async_to_lds_b128 %0, %1, off"
                     :
                     : "v"(lds_base + i * 4096), "v"(gsrc + i * 1024)
                     : "memory");
      }
    }
    wait_async0();
    __syncthreads();                              // chunk visible to all waves

#pragma unroll 4
    for (int k = 0; k < KCHUNK; k += 4) {
      v2f a, b;
      a.x = arow[kc + k];
      a.y = arow[kc + k + 1];
      b.x = bcol[(size_t)k * OUT_DIM];            // ds_load from staged B
      b.y = bcol[(size_t)(k + 1) * OUT_DIM];
      acc = __builtin_amdgcn_wmma_f32_16x16x4_f32(
          /*neg_a=*/false, a, /*neg_b=*/false, b,
          /*c_mod=*/(short)0, acc, /*reuse_a=*/false, /*reuse_b=*/false);
    }
  }

  const float bv = bias ? bias[n0 + l16] : 0.0f;
#pragma unroll
  for (int r = 0; r < 8; ++r) {
    const int m = m0 + r + 8 * half;
    C[(size_t)m * ldc + n0 + l16] = acc[r] + bv;
  }
}

// ---------------- edge kernel: one wave32 per edge --------------------------
// score = cos(a_tab[src], b_tab[dst]); scatter score * m_tab[src] into agg[dst].
// cnt/ssum (may be null) accumulate per-dst count and score sum (mean attn).
__global__ void __launch_bounds__(256)
edge_scatter(const float* __restrict__ a_tab,            // fresh hr, stride 128
             const float* __restrict__ m_tab,            // hr_final, stride 128
             const float* __restrict__ b_tab, int ldb,   // ht (dst side)
             const int* __restrict__ src, const int* __restrict__ dst,
             float* __restrict__ agg, int ldagg,         // accumulate target
             float* __restrict__ cnt, float* __restrict__ ssum,
             int nE)
{
  const int e    = (int)(((size_t)blockIdx.x * blockDim.x + threadIdx.x) >> 5);
  const int lane = threadIdx.x & 31;
  if (e >= nE) return;

  const int s = src[e];
  const int d = dst[e];

  const float4 av = ((const float4*)(a_tab + (size_t)s * OUT_DIM))[lane];
  const float4 bv = ((const float4*)(b_tab + (size_t)d * ldb))[lane];
  // issue the message gather early; overlaps the shuffle-reduce chain
  const float4 mv = ((const float4*)(m_tab + (size_t)s * OUT_DIM))[lane];

  float dot = av.x * bv.x + av.y * bv.y + av.z * bv.z + av.w * bv.w;
  float na  = av.x * av.x + av.y * av.y + av.z * av.z + av.w * av.w;
  float nb  = bv.x * bv.x + bv.y * bv.y + bv.z * bv.z + bv.w * bv.w;

#pragma unroll
  for (int off = 16; off > 0; off >>= 1) {
    dot += __shfl_xor(dot, off, 32);
    na  += __shfl_xor(na,  off, 32);
    nb  += __shfl_xor(nb,  off, 32);
  }

  const float sc = dot / (fmaxf(sqrtf(na), 1e-8f) * fmaxf(sqrtf(nb), 1e-8f));

  float* o = agg + (size_t)d * ldagg + lane * 4;
  atomicAdd(o + 0, sc * mv.x);
  atomicAdd(o + 1, sc * mv.y);
  atomicAdd(o + 2, sc * mv.z);
  atomicAdd(o + 3, sc * mv.w);

  if (cnt != nullptr && lane == 0) {
    atomicAdd(cnt  + d, 1.0f);
    atomicAdd(ssum + d, sc);
  }
}

// ---------------- vul combine: pairwise softmax (0.6 pos / 0.4 neg) ---------
__global__ void __launch_bounds__(256)
combine_vul(const float* __restrict__ agg,    // 4 blocks of [N_VUL,128]
            const float* __restrict__ cnt,    // 4 blocks of [N_VUL]
            const float* __restrict__ ssum,
            float* __restrict__ out0)         // d_out, row stride 256
{
  const int idx = blockIdx.x * blockDim.x + threadIdx.x;
  if (idx >= N_VUL * OUT_DIM) return;
  const int row = idx >> 7;
  const int col = idx & (OUT_DIM - 1);

  float m[4];
#pragma unroll
  for (int i = 0; i < 4; ++i)
    m[i] = ssum[(size_t)i * N_VUL + row] /
           fmaxf(cnt[(size_t)i * N_VUL + row], 1.0f);

  const float mp = fmaxf(m[0], m[1]);
  const float e0 = __expf(m[0] - mp), e1 = __expf(m[1] - mp);
  const float wp = 0.6f / (e0 + e1);
  const float mn = fmaxf(m[2], m[3]);
  const float e2 = __expf(m[2] - mn), e3 = __expf(m[3] - mn);
  const float wn = 0.4f / (e2 + e3);

  const size_t off = (size_t)row * OUT_DIM + col;
  const size_t blk = (size_t)N_VUL * OUT_DIM;
  const float h = wp * (e0 * agg[off] + e1 * agg[blk + off]) +
                  wn * (e2 * agg[2 * blk + off] + e3 * agg[3 * blk + off]);

  out0[(size_t)row * 256 + OUT_DIM + col] = h;
}

// ---------------------------------------------------------------------------
extern "C" void kernel_launch(void* const* d_in, const int* in_sizes, int n_in,
                              void* d_out, int out_size, void* d_ws, size_t ws_size,
                              hipStream_t stream)
{
  const float* feat_vul = (const float*)d_in[0];  // [100000,256]
  const float* feat_oth = (const float*)d_in[1];  // [4,20000,256]
  const float* We       = (const float*)d_in[2];  // [8,256,128]
  const float* Wn       = (const float*)d_in[3];  // [5,256,128]
  const float* bn       = (const float*)d_in[4];  // [5,128]
  const int*   src      = (const int*)d_in[5];    // [8,400000]
  const int*   dst      = (const int*)d_in[6];    // [8,400000]
  float* out = (float*)d_out;

  // workspace layout (floats):
  float* ws     = (float*)d_ws;
  float* hr_oth = ws;                                        //  4*20000*128
  float* hr_vul = hr_oth + (size_t)4 * N_OTH * OUT_DIM;      //  100000*128
  float* hr_tmp = hr_vul + (size_t)N_VUL * OUT_DIM;          //  100000*128
  float* agg    = hr_tmp + (size_t)N_VUL * OUT_DIM;          //  4*100000*128
  float* cnt    = agg    + (size_t)4 * N_VUL * OUT_DIM;      //  4*100000
  float* ssum   = cnt    + (size_t)4 * N_VUL;                //  4*100000

  float* out_n[5];
  out_n[0] = out;                                            // [100000,256]
  for (int n = 1; n < 5; ++n)
    out_n[n] = out + (size_t)N_VUL * 256 + (size_t)(n - 1) * N_OTH * 256;

  // 1) zero accumulators (ws) and the 4 non-vul output blocks (their
  //    cols 128..255 receive atomic message sums; cols 0..127 are
  //    overwritten by the ht GEMMs below).
  zero_f32<<<2048, 256, 0, stream>>>(agg, (size_t)4 * N_VUL * OUT_DIM + (size_t)8 * N_VUL);
  zero_f32<<<2048, 256, 0, stream>>>(out_n[1], (size_t)4 * N_OTH * 256);

  // 2) ht[n] = feats[n] @ Wn[n] + bn[n]  -> d_out columns 0..127
  wmma_gemm_f32<<<N_VUL / 16, 256, 0, stream>>>(feat_vul, Wn, bn, out_n[0], 256, N_VUL);
  for (int n = 1; n < 5; ++n)
    wmma_gemm_f32<<<N_OTH / 16, 256, 0, stream>>>(
        feat_oth + (size_t)(n - 1) * N_OTH * IN_DIM,
        Wn + (size_t)n * IN_DIM * OUT_DIM,
        bn + (size_t)n * OUT_DIM, out_n[n], 256, N_OTH);

  // 3) hr_final for src types 1..4 (etypes 0..3, fresh == final) and for
  //    vul (last write in ref loop is etype 7 -> We[7]).
  for (int k = 0; k < 4; ++k)
    wmma_gemm_f32<<<N_OTH / 16, 256, 0, stream>>>(
        feat_oth + (size_t)k * N_OTH * IN_DIM,
        We + (size_t)k * IN_DIM * OUT_DIM,
        nullptr, hr_oth + (size_t)k * N_OTH * OUT_DIM, OUT_DIM, N_OTH);
  wmma_gemm_f32<<<N_VUL / 16, 256, 0, stream>>>(
      feat_vul, We + (size_t)7 * IN_DIM * OUT_DIM, nullptr, hr_vul, OUT_DIM, N_VUL);

  const int EBLK = (NE + 7) / 8;  // one wave per edge, 8 waves / 256-thr block

  // 4) etypes 0..3: other -> vul; messages need cnt/ssum for mean attention.
  for (int i = 0; i < 4; ++i)
    edge_scatter<<<EBLK, 256, 0, stream>>>(
        hr_oth + (size_t)i * N_OTH * OUT_DIM,
        hr_oth + (size_t)i * N_OTH * OUT_DIM,
        out_n[0], 256,
        src + (size_t)i * NE, dst + (size_t)i * NE,
        agg + (size_t)i * N_VUL * OUT_DIM, OUT_DIM,
        cnt + (size_t)i * N_VUL, ssum + (size_t)i * N_VUL, NE);

  // 5) etypes 4..6: vul -> other; scores use fresh hr (We[i]), messages use
  //    hr_final[0] (We[7]); size-1 softmax => scatter straight into d_out.
  for (int i = 4; i < 7; ++i) {
    wmma_gemm_f32<<<N_VUL / 16, 256, 0, stream>>>(
        feat_vul, We + (size_t)i * IN_DIM * OUT_DIM, nullptr, hr_tmp, OUT_DIM, N_VUL);
    edge_scatter<<<EBLK, 256, 0, stream>>>(
        hr_tmp, hr_vul, out_n[i - 3], 256,
        src + (size_t)i * NE, dst + (size_t)i * NE,
        out_n[i - 3] + OUT_DIM, 256, nullptr, nullptr, NE);
  }
  // etype 7: fresh hr == hr_final[0]
  edge_scatter<<<EBLK, 256, 0, stream>>>(
      hr_vul, hr_vul, out_n[4], 256,
      src + (size_t)7 * NE, dst + (size_t)7 * NE,
      out_n[4] + OUT_DIM, 256, nullptr, nullptr, NE);

  // 6) vul combine: softmax over (w2v,p2v)*0.6 + (r2v,d2v)*0.4
  combine_vul<<<(N_VUL * OUT_DIM + 255) / 256, 256, 0, stream>>>(agg, cnt, ssum, out_n[0]);
}